// Coords2RMSD_26671746908331
// MI455X (gfx1250) — compile-verified
//
#include <hip/hip_runtime.h>
#include <stdint.h>

#define NMAX        2048
#define TPB         256
#define TILE_ATOMS  256
#define TILES       (NMAX / TILE_ATOMS)        // 8
#define TILE_FLOATS (TILE_ATOMS * 3)           // 768
#define TILE_BYTES  (TILE_FLOATS * 4)          // 3072
#define BUF_FLOATS  (TILE_FLOATS * 2)          // 1536 floats: x-tile then y-tile
#define BUF_BYTES   (TILE_BYTES * 2)           // 6144

#if defined(__has_builtin)
#if __has_builtin(__builtin_amdgcn_wmma_f32_16x16x4_f32)
#define HAVE_WMMA_F32X4 1
#endif
#endif
#ifndef HAVE_WMMA_F32X4
#define HAVE_WMMA_F32X4 0
#endif

typedef float v2f __attribute__((ext_vector_type(2)));
typedef float v8f __attribute__((ext_vector_type(8)));

// One lane moves 16 bytes global -> LDS via the CDNA5 async engine (ASYNCcnt).
__device__ __forceinline__ void async_ld16(uint32_t lds_off, const void* g) {
    uint64_t ga = (uint64_t)(uintptr_t)g;
    asm volatile("global_load_async_to_lds_b128 %0, %1, off"
                 :: "v"(lds_off), "v"(ga) : "memory");
}

__device__ __forceinline__ void wait_async_le2() {
#if defined(__has_builtin) && __has_builtin(__builtin_amdgcn_s_wait_asynccnt)
    __builtin_amdgcn_s_wait_asynccnt(2);
#else
    asm volatile("s_wait_asynccnt 2" ::: "memory");
#endif
}
__device__ __forceinline__ void wait_async_le0() {
#if defined(__has_builtin) && __has_builtin(__builtin_amdgcn_s_wait_asynccnt)
    __builtin_amdgcn_s_wait_asynccnt(0);
#else
    asm volatile("s_wait_asynccnt 0" ::: "memory");
#endif
}

__global__ __launch_bounds__(TPB) void kabsch_rmsd_kernel(
    const float* __restrict__ xg, const float* __restrict__ yg,
    const int* __restrict__ natoms, float* __restrict__ out) {

    __shared__ float stage[2 * BUF_FLOATS]; // 12288 B: double-buffered x|y tiles
    __shared__ float dred[8 * 64];          // 2048 B: per-wave 8x8 D blocks

    const int b    = blockIdx.x;
    const int tid  = threadIdx.x;
    const int lane = tid & 31;
    const int wv   = tid >> 5;
    const int nat  = natoms[b];

    const char* xb = (const char*)(xg + (size_t)b * (3 * NMAX));
    const char* yb = (const char*)(yg + (size_t)b * (3 * NMAX));
    const uint32_t stage_lds = (uint32_t)(uintptr_t)stage; // low 32 bits of flat = LDS offset
    const char* sbase = (const char*)stage;                // LDS byte base for ds loads

    // Prefetch tile 0 into buffer 0 (192 lanes x 16B = 3072B per array)
    if (tid < TILE_BYTES / 16) {
        const uint32_t o = (uint32_t)tid * 16u;
        async_ld16(stage_lds + o,              xb + o);
        async_ld16(stage_lds + TILE_BYTES + o, yb + o);
    }

#if HAVE_WMMA_F32X4
    v8f d = {};                     // per-wave accumulator: D = A x B + D
    const int r  = lane & 15;       // A row / B col handled by this lane
    const int hi = lane >> 4;       // upper lane half holds K=2,3

    // Branchless per-lane fragment recipe (loop-invariant):
    // A row r of atom e = [x0,x1,x2, 1, y0,y1,y2, 0...];  B col swaps x<->y.
    // value = (ds_load(off) * wsel + csel) * validmask
    int offA = 0, offB = 0;         // byte offset within atom record (+ x/y region)
    float wsel = 0.f, csel = 0.f;
    if (r < 3)       { offA = r * 4;                    offB = TILE_BYTES + r * 4;  wsel = 1.f; }
    else if (r == 3) { csel = 1.f; }
    else if (r < 7)  { offA = TILE_BYTES + (r - 4) * 4; offB = (r - 4) * 4;         wsel = 1.f; }
    // rows/cols 7..15: offs 0, wsel = csel = 0 -> contribute zero
#else
    float sx0=0.f,sx1=0.f,sx2=0.f, sy0=0.f,sy1=0.f,sy2=0.f;
    float c00=0.f,c01=0.f,c02=0.f, c10=0.f,c11=0.f,c12=0.f, c20=0.f,c21=0.f,c22=0.f;
    float nxx=0.f, nyy=0.f;
#endif

    for (int t = 0; t < TILES; ++t) {
        const int buf = t & 1;
        if (t + 1 < TILES) {
            if (tid < TILE_BYTES / 16) {
                const uint32_t o    = (uint32_t)tid * 16u;
                const uint32_t dst  = stage_lds + (uint32_t)(buf ^ 1) * BUF_BYTES + o;
                const uint32_t goff = (uint32_t)(t + 1) * TILE_BYTES + o;
                async_ld16(dst,              xb + goff);
                async_ld16(dst + TILE_BYTES, yb + goff);
            }
            wait_async_le2();   // tile t's two async ops (oldest) are complete
        } else {
            wait_async_le0();
        }
        __syncthreads();        // tile t visible workgroup-wide

#if HAVE_WMMA_F32X4
        // Wave wv consumes tile atoms [wv*32, wv*32+32) in 8 WMMA steps of K=4.
        // A 16x4 f32 layout: VGPR0 = {K0 | K2}, VGPR1 = {K1 | K3}; B 4x16 mirrors.
        const int bufbase = buf * BUF_BYTES;
        #pragma unroll
        for (int step = 0; step < 8; ++step) {
            const int e0 = wv * 32 + step * 4 + (hi ? 2 : 0); // atom for VGPR0
            const int g0 = t * TILE_ATOMS + e0;
            const float m0 = (g0     < nat) ? 1.f : 0.f;      // data masking, EXEC stays all-1s
            const float m1 = (g0 + 1 < nat) ? 1.f : 0.f;
            const int byte0 = bufbase + 12 * e0;
            const int byte1 = byte0 + 12;                     // atom e0+1
            const float la0 = *(const float*)(sbase + byte0 + offA);
            const float la1 = *(const float*)(sbase + byte1 + offA);
            const float lb0 = *(const float*)(sbase + byte0 + offB);
            const float lb1 = *(const float*)(sbase + byte1 + offB);
            v2f A, Bv;
            A[0]  = (la0 * wsel + csel) * m0;
            A[1]  = (la1 * wsel + csel) * m1;
            Bv[0] = (lb0 * wsel + csel) * m0;
            Bv[1] = (lb1 * wsel + csel) * m1;
            d = __builtin_amdgcn_wmma_f32_16x16x4_f32(
                    false, A, false, Bv, (short)0, d, false, false);
        }
#else
        const float* lx = &stage[buf * BUF_FLOATS];
        const float* ly = lx + TILE_FLOATS;
        const int a = t * TILE_ATOMS + tid;
        const float x0 = lx[3*tid+0], x1 = lx[3*tid+1], x2 = lx[3*tid+2];
        const float y0 = ly[3*tid+0], y1 = ly[3*tid+1], y2 = ly[3*tid+2];
        if (a < nat) {
            sx0 += x0; sx1 += x1; sx2 += x2;
            sy0 += y0; sy1 += y1; sy2 += y2;
            c00 += x0*y0; c01 += x0*y1; c02 += x0*y2;
            c10 += x1*y0; c11 += x1*y1; c12 += x1*y2;
            c20 += x2*y0; c21 += x2*y1; c22 += x2*y2;
            nxx += x0*x0 + x1*x1 + x2*x2;
            nyy += y0*y0 + y1*y1 + y2*y2;
        }
#endif
        __syncthreads();        // everyone done reading buf before it is overwritten
    }

#if HAVE_WMMA_F32X4
    // C/D layout: VGPR m, lanes 0-15 hold D[m][lane]. Need m,n in 0..7.
    if (lane < 8) {
        #pragma unroll
        for (int m = 0; m < 8; ++m) dred[wv * 64 + m * 8 + lane] = d[m];
    }
    __syncthreads();
    if (tid < 64) {             // tid = m*8+n: sum the 8 per-wave D blocks
        float s = 0.f;
        #pragma unroll
        for (int w = 0; w < 8; ++w) s += dred[w * 64 + tid];
        dred[tid] = s;
    }
    __syncthreads();
#else
    __shared__ float red[TPB * 18];
    {
        float acc[17] = {sx0,sx1,sx2, sy0,sy1,sy2,
                         c00,c01,c02, c10,c11,c12, c20,c21,c22, nxx,nyy};
        #pragma unroll
        for (int k = 0; k < 17; ++k) red[tid*18 + k] = acc[k];
    }
    __syncthreads();
    for (int s = TPB / 2; s > 0; s >>= 1) {
        if (tid < s) {
            #pragma unroll
            for (int k = 0; k < 17; ++k) red[tid*18 + k] += red[(tid+s)*18 + k];
        }
        __syncthreads();
    }
#endif

    if (tid == 0) {
        const float n = (float)nat, invn = 1.f / n;
#if HAVE_WMMA_F32X4
        #define DF(m, nn) dred[(m)*8 + (nn)]
        const float Sx0 = DF(0,3), Sx1 = DF(1,3), Sx2 = DF(2,3);
        const float Sy0 = DF(3,0), Sy1 = DF(3,1), Sy2 = DF(3,2);
        const float r00 = DF(0,0), r01 = DF(0,1), r02 = DF(0,2);
        const float r10 = DF(1,0), r11 = DF(1,1), r12 = DF(1,2);
        const float r20 = DF(2,0), r21 = DF(2,1), r22 = DF(2,2);
        const float sumxx = DF(0,4) + DF(1,5) + DF(2,6);   // Sum|x|^2
        const float sumyy = DF(4,0) + DF(5,1) + DF(6,2);   // Sum|y|^2
        #undef DF
#else
        const float Sx0=red[0], Sx1=red[1], Sx2=red[2];
        const float Sy0=red[3], Sy1=red[4], Sy2=red[5];
        const float r00=red[6],  r01=red[7],  r02=red[8];
        const float r10=red[9],  r11=red[10], r12=red[11];
        const float r20=red[12], r21=red[13], r22=red[14];
        const float sumxx=red[15], sumyy=red[16];
#endif
        // Centered cross-covariance: C = Sum(x y^T) - Sx Sy^T / n
        const float C00=r00-Sx0*Sy0*invn, C01=r01-Sx0*Sy1*invn, C02=r02-Sx0*Sy2*invn;
        const float C10=r10-Sx1*Sy0*invn, C11=r11-Sx1*Sy1*invn, C12=r12-Sx1*Sy2*invn;
        const float C20=r20-Sx2*Sy0*invn, C21=r21-Sx2*Sy1*invn, C22=r22-Sx2*Sy2*invn;
        const float gx = sumxx - (Sx0*Sx0+Sx1*Sx1+Sx2*Sx2)*invn;
        const float gy = sumyy - (Sy0*Sy0+Sy1*Sy1+Sy2*Sy2)*invn;

        const float det = C00*(C11*C22 - C12*C21)
                        - C01*(C10*C22 - C12*C20)
                        + C02*(C10*C21 - C11*C20);

        // Bm = C^T C (symmetric PSD); singular values of C = sqrt(eigvals(Bm))
        const float b00 = C00*C00 + C10*C10 + C20*C20;
        const float b11 = C01*C01 + C11*C11 + C21*C21;
        const float b22 = C02*C02 + C12*C12 + C22*C22;
        const float b01 = C00*C01 + C10*C11 + C20*C21;
        const float b02 = C00*C02 + C10*C12 + C20*C22;
        const float b12 = C01*C02 + C11*C12 + C21*C22;

        const float q  = (b00 + b11 + b22) * (1.f/3.f);
        const float p1 = b01*b01 + b02*b02 + b12*b12;
        const float a00 = b00 - q, a11 = b11 - q, a22 = b22 - q;
        const float p2 = a00*a00 + a11*a11 + a22*a22 + 2.f*p1;

        float l0, l1, l2;
        if (p2 > 1e-30f) {
            const float p    = sqrtf(p2 * (1.f/6.f));
            const float invp = 1.f / p;
            const float dA = a00*(a11*a22 - b12*b12)
                           - b01*(b01*a22 - b12*b02)
                           + b02*(b01*b12 - a11*b02);
            float rr = 0.5f * dA * invp*invp*invp;
            rr = fminf(1.f, fmaxf(-1.f, rr));
            const float phi = acosf(rr) * (1.f/3.f);
            l0 = q + 2.f*p*cosf(phi);
            l2 = q + 2.f*p*cosf(phi + 2.0943951023931953f); // +2*pi/3
            l1 = 3.f*q - l0 - l2;
        } else {
            l0 = l1 = l2 = q;
        }
        l0 = fmaxf(l0, 0.f); l1 = fmaxf(l1, 0.f); l2 = fmaxf(l2, 0.f);
        const float s2 = sqrtf(l2);
        const float tr = sqrtf(l0) + sqrtf(l1) + (det < 0.f ? -s2 : s2);
        const float msd = (gx + gy - 2.f*tr) * invn;
        out[b] = sqrtf(fmaxf(msd, 0.f));
    }
}

extern "C" void kernel_launch(void* const* d_in, const int* in_sizes, int n_in,
                              void* d_out, int out_size, void* d_ws, size_t ws_size,
                              hipStream_t stream) {
    (void)n_in; (void)d_ws; (void)ws_size; (void)out_size;
    const float* x   = (const float*)d_in[0];
    const float* y   = (const float*)d_in[1];
    const int*   nat = (const int*)d_in[2];
    float*       out = (float*)d_out;
    const int B = in_sizes[2]; // 4096
    kabsch_rmsd_kernel<<<B, TPB, 0, stream>>>(x, y, nat, out);
}